// Lupool_66915590471830
// MI455X (gfx1250) — compile-verified
//
#include <hip/hip_runtime.h>
#include <stdint.h>

typedef float v2f __attribute__((ext_vector_type(2)));
typedef float v8f __attribute__((ext_vector_type(8)));

#define CDIV(a, b) (((a) + (b) - 1) / (b))

// ---------------------------------------------------------------------------
// 1) degrees (float atomic counts; exact for counts < 2^24)
// ---------------------------------------------------------------------------
__global__ void k_degree(const long long* __restrict__ ei,
                         float* __restrict__ deg_out, float* __restrict__ deg_in,
                         int E) {
  int e = blockIdx.x * blockDim.x + threadIdx.x;
  if (e >= E) return;
  int s = (int)ei[e];
  int d = (int)ei[(long long)E + e];
  atomicAdd(&deg_out[s], 1.0f);
  atomicAdd(&deg_in[d], 1.0f);
}

__global__ void k_norm(const float* __restrict__ deg_out, const float* __restrict__ deg_in,
                       float* __restrict__ nsrc, float* __restrict__ ndst, int N) {
  int i = blockIdx.x * blockDim.x + threadIdx.x;
  if (i >= N) return;
  nsrc[i] = rsqrtf(fmaxf(deg_out[i], 1.0f));
  ndst[i] = rsqrtf(fmaxf(deg_in[i], 1.0f));
}

// ---------------------------------------------------------------------------
// 2) layer-1 edge aggregation: agg[dst] += x[src] * nsrc[src]   (128 wide)
//    one wave32 per edge, 4 channels per lane (float4 load, 4 f32 atomics)
// ---------------------------------------------------------------------------
__global__ void k_scatter1(const long long* __restrict__ ei,
                           const float* __restrict__ x,
                           const float* __restrict__ nsrc,
                           float* __restrict__ agg, int E) {
  int wid  = (blockIdx.x * blockDim.x + threadIdx.x) >> 5;
  int lane = threadIdx.x & 31;
  if (wid >= E) return;
  long long s = ei[wid];
  long long d = ei[(long long)E + wid];
  float ns = nsrc[s];
  const float4* xs = (const float4*)(x + s * 128);
  float4 v = xs[lane];
  float* ad = agg + d * 128 + lane * 4;
  atomicAdd(ad + 0, v.x * ns);
  atomicAdd(ad + 1, v.y * ns);
  atomicAdd(ad + 2, v.z * ns);
  atomicAdd(ad + 3, v.w * ns);
}

// ---------------------------------------------------------------------------
// 3) GEMM1 + bias + relu using V_WMMA_F32_16X16X4_F32.
//    score[N,192] = relu( (agg * ndst[:,None]) @ W1 + b1 )
//    One wave computes one 16x16 tile; K=128 stepped by 4.
//    A layout: lanes 0-15 -> (M=lane, K=k0..k0+1); lanes 16-31 -> (M=lane-16, K=k0+2..k0+3)
//    B layout: mirrored with N=lane&15.  C/D: VGPR r -> M = r + 8*(lane>=16), N = lane&15.
// ---------------------------------------------------------------------------
__global__ void k_gemm1(const float* __restrict__ agg,
                        const float* __restrict__ ndst,
                        const float* __restrict__ W1,
                        const float* __restrict__ b1,
                        float* __restrict__ score,
                        int N, int Mtiles) {
  int wid  = (blockIdx.x * blockDim.x + threadIdx.x) >> 5;
  int lane = threadIdx.x & 31;
  int mt = wid / 12;          // 192/16 = 12 N-tiles
  int nt = wid % 12;
  if (mt >= Mtiles) return;   // wave-uniform; skipped via execz branch

  int half = lane >> 4;       // 0: K+0..1, 1: K+2..3
  int l    = lane & 15;
  int m    = mt * 16 + l;
  int n    = nt * 16 + l;
  int mld  = m < N ? m : N - 1;   // clamp loads for tail tiles (keeps EXEC all-1)
  float nd = ndst[mld];
  const float* Arow = agg + (size_t)mld * 128;

  v8f c = {};
  for (int k0 = 0; k0 < 128; k0 += 4) {
    int kb = k0 + 2 * half;
    v2f a;
    a.x = Arow[kb] * nd;
    a.y = Arow[kb + 1] * nd;
    v2f b;
    b.x = W1[(size_t)kb * 192 + n];
    b.y = W1[(size_t)(kb + 1) * 192 + n];
    c = __builtin_amdgcn_wmma_f32_16x16x4_f32(false, a, false, b, (short)0, c,
                                              false, false);
  }
  float bias = b1[n];
  float* srow = score + (size_t)(mt * 16 + 8 * half) * 192 + n;
  if (mt * 16 + 16 <= N) {
    // full tile (wave-uniform): straight-line stores, no per-element masking
#pragma unroll
    for (int r = 0; r < 8; ++r) {
      float v = c[r] + bias;
      srow[(size_t)r * 192] = v > 0.0f ? v : 0.0f;
    }
  } else {
#pragma unroll
    for (int r = 0; r < 8; ++r) {
      int mm = mt * 16 + r + 8 * half;
      if (mm < N) {
        float v = c[r] + bias;
        score[(size_t)mm * 192 + n] = v > 0.0f ? v : 0.0f;
      }
    }
  }
}

// ---------------------------------------------------------------------------
// 4) s1[u] = nsrc[u] * dot(score[u,:], W2)   (one wave per node, shuffle reduce)
// ---------------------------------------------------------------------------
__global__ void k_s1(const float* __restrict__ score, const float* __restrict__ W2,
                     const float* __restrict__ nsrc, float* __restrict__ s1, int N) {
  int wid  = (blockIdx.x * blockDim.x + threadIdx.x) >> 5;
  int lane = threadIdx.x & 31;
  if (wid >= N) return;
  const float* sr = score + (size_t)wid * 192;
  float acc = 0.0f;
  for (int j = lane; j < 192; j += 32) acc += sr[j] * W2[j];
  for (int off = 16; off > 0; off >>= 1) acc += __shfl_down(acc, off, 32);
  if (lane == 0) s1[wid] = acc * nsrc[wid];
}

// 5) layer-2 edge aggregation collapsed to scalar: sacc[dst] += s1[src]
__global__ void k_scatter2(const long long* __restrict__ ei,
                           const float* __restrict__ s1,
                           float* __restrict__ sacc, int E) {
  int e = blockIdx.x * blockDim.x + threadIdx.x;
  if (e >= E) return;
  long long s = ei[e];
  long long d = ei[(long long)E + e];
  atomicAdd(&sacc[d], s1[s]);
}

// 6) keys[i] = float_bits(|ndst*sacc + b2|)  (monotone for non-negative floats)
__global__ void k_keys(const float* __restrict__ sacc, const float* __restrict__ ndst,
                       const float* __restrict__ b2, unsigned* __restrict__ keys, int N) {
  int i = blockIdx.x * blockDim.x + threadIdx.x;
  if (i >= N) return;
  float sc = ndst[i] * sacc[i] + b2[0];
  keys[i] = __float_as_uint(fabsf(sc));
}

// ---------------------------------------------------------------------------
// 7) exact top-k threshold: 2-level 16-bit radix select
// ---------------------------------------------------------------------------
__global__ void k_hist_hi(const unsigned* __restrict__ keys, unsigned* __restrict__ hist, int N) {
  int i = blockIdx.x * blockDim.x + threadIdx.x;
  if (i >= N) return;
  atomicAdd(&hist[keys[i] >> 16], 1u);
}

__device__ void select_from_hist(const unsigned* __restrict__ hist, unsigned k,
                                 unsigned* outBin, unsigned* outRem) {
  __shared__ unsigned psum[1024];
  int t = threadIdx.x;
  const int chunk = 65536 / 1024;  // 64
  unsigned s = 0;
  int base = t * chunk;
  for (int j = 0; j < chunk; ++j) s += hist[base + j];
  psum[t] = s;
  __syncthreads();
  if (t == 0) {
    unsigned cum = 0;
    for (int ci = 1023; ci >= 0; --ci) {
      unsigned cs = psum[ci];
      if (cum + cs >= k) {
        int b0 = ci * chunk;
        for (int b = b0 + chunk - 1; b >= b0; --b) {
          unsigned h = hist[b];
          if (cum + h >= k) { *outBin = (unsigned)b; *outRem = k - cum; return; }
          cum += h;
        }
      }
      cum += cs;
    }
    *outBin = 0; *outRem = k - cum;
  }
}

__global__ void k_sel1(const unsigned* __restrict__ hist, unsigned k, unsigned* __restrict__ ctrl) {
  select_from_hist(hist, k, &ctrl[0], &ctrl[1]);  // ctrl[0]=hi bin, ctrl[1]=remaining
}

__global__ void k_hist_lo(const unsigned* __restrict__ keys, const unsigned* __restrict__ ctrl,
                          unsigned* __restrict__ hist2, int N) {
  int i = blockIdx.x * blockDim.x + threadIdx.x;
  if (i >= N) return;
  unsigned kk = keys[i];
  if ((kk >> 16) == ctrl[0]) atomicAdd(&hist2[kk & 0xFFFFu], 1u);
}

__global__ void k_sel2(const unsigned* __restrict__ hist2, unsigned* __restrict__ ctrl) {
  __shared__ unsigned binLo, rem;
  select_from_hist(hist2, ctrl[1], &binLo, &rem);
  if (threadIdx.x == 0) {
    ctrl[2] = (ctrl[0] << 16) | binLo;  // exact 32-bit threshold T
    ctrl[3] = rem;                      // tie quota at key == T
  }
}

// ---------------------------------------------------------------------------
// 8) deterministic stable compaction: dual prefix scan packed in u64
//    high 32: count(key > T), low 32: count(key == T)
// ---------------------------------------------------------------------------
__global__ void k_scan1(const unsigned* __restrict__ keys, const unsigned* __restrict__ ctrl,
                        unsigned long long* __restrict__ pref,
                        unsigned long long* __restrict__ btot, int N) {
  __shared__ unsigned long long sh[256];
  int t = threadIdx.x;
  int base = blockIdx.x * 1024 + t * 4;
  unsigned T = ctrl[2];
  unsigned long long v[4];
  unsigned long long s = 0;
  for (int j = 0; j < 4; ++j) {
    int i = base + j;
    unsigned long long e = 0;
    if (i < N) {
      unsigned kk = keys[i];
      e = (kk > T) ? (1ULL << 32) : (kk == T ? 1ULL : 0ULL);
    }
    v[j] = e; s += e;
  }
  sh[t] = s;
  __syncthreads();
  for (int off = 1; off < 256; off <<= 1) {
    unsigned long long xv = (t >= off) ? sh[t - off] : 0ULL;
    __syncthreads();
    sh[t] += xv;
    __syncthreads();
  }
  unsigned long long excl = (t == 0) ? 0ULL : sh[t - 1];
  if (t == 255) btot[blockIdx.x] = sh[255];
  for (int j = 0; j < 4; ++j) {
    int i = base + j;
    if (i < N) pref[i] = excl;
    excl += v[j];
  }
}

__global__ void k_scan2(unsigned long long* __restrict__ btot, int nb) {
  if (blockIdx.x == 0 && threadIdx.x == 0) {
    unsigned long long run = 0;
    for (int i = 0; i < nb; ++i) { unsigned long long t = btot[i]; btot[i] = run; run += t; }
  }
}

__global__ void k_emit(const unsigned* __restrict__ keys, const unsigned* __restrict__ ctrl,
                       const unsigned long long* __restrict__ pref,
                       const unsigned long long* __restrict__ btot,
                       unsigned* __restrict__ kept, int N) {
  int i = blockIdx.x * blockDim.x + threadIdx.x;
  if (i >= N) return;
  unsigned T = ctrl[2], quota = ctrl[3];
  unsigned kk = keys[i];
  unsigned long long p = pref[i] + btot[i >> 10];
  unsigned prefA = (unsigned)(p >> 32);          // # key>T before i
  unsigned prefB = (unsigned)(p & 0xFFFFFFFFu);  // # key==T before i
  bool sel = (kk > T) || (kk == T && prefB < quota);
  if (sel) {
    unsigned tiesBefore = prefB < quota ? prefB : quota;
    kept[prefA + tiesBefore] = (unsigned)i;
  }
}

// 9) gather kept rows (float4, one wave per row)
__global__ void k_gather(const float* __restrict__ x, const unsigned* __restrict__ kept,
                         float* __restrict__ out, int K) {
  int tid  = blockIdx.x * blockDim.x + threadIdx.x;
  int row  = tid >> 5;
  int lane = tid & 31;
  if (row >= K) return;
  unsigned src = kept[row];
  const float4* xs = (const float4*)(x + (size_t)src * 128);
  float4* od = (float4*)(out + (size_t)row * 128);
  od[lane] = xs[lane];
}

// ---------------------------------------------------------------------------
extern "C" void kernel_launch(void* const* d_in, const int* in_sizes, int n_in,
                              void* d_out, int out_size, void* d_ws, size_t ws_size,
                              hipStream_t stream) {
  const float*     x   = (const float*)d_in[0];
  const long long* ei  = (const long long*)d_in[1];
  const float*     W1  = (const float*)d_in[2];
  const float*     b1  = (const float*)d_in[3];
  const float*     W2  = (const float*)d_in[4];
  const float*     b2  = (const float*)d_in[5];
  float* out = (float*)d_out;

  const int N = in_sizes[0] / 128;
  const int E = in_sizes[1] / 2;
  const int K = out_size / 128;

  // ---- workspace bump allocator (zero-region first, single memset) ----
  char* ws = (char*)d_ws;
  size_t off = 0;
  auto alloc = [&](size_t bytes) -> char* {
    char* p = ws + off;
    off = (off + bytes + 255) & ~(size_t)255;
    return p;
  };
  float*    deg_out = (float*)alloc((size_t)N * 4);
  float*    deg_in  = (float*)alloc((size_t)N * 4);
  float*    sacc    = (float*)alloc((size_t)N * 4);
  unsigned* hist1   = (unsigned*)alloc(65536 * 4);
  unsigned* hist2   = (unsigned*)alloc(65536 * 4);
  unsigned* ctrl    = (unsigned*)alloc(64);
  float*    agg     = (float*)alloc((size_t)N * 128 * 4);
  size_t zero_bytes = off;                          // everything above must start at 0
  float*    nsrc    = (float*)alloc((size_t)N * 4);
  float*    ndst    = (float*)alloc((size_t)N * 4);
  float*    score   = (float*)alloc((size_t)N * 192 * 4);
  float*    s1      = (float*)alloc((size_t)N * 4);
  unsigned* keys    = (unsigned*)alloc((size_t)N * 4);
  unsigned long long* pref = (unsigned long long*)alloc((size_t)N * 8);
  unsigned long long* btot = (unsigned long long*)alloc((size_t)CDIV(N, 1024) * 8 + 256);
  unsigned* kept    = (unsigned*)alloc((size_t)K * 4);
  (void)ws_size;

  hipMemsetAsync(d_ws, 0, zero_bytes, stream);

  const int B = 256;
  // degrees + norms
  k_degree<<<CDIV(E, B), B, 0, stream>>>(ei, deg_out, deg_in, E);
  k_norm<<<CDIV(N, B), B, 0, stream>>>(deg_out, deg_in, nsrc, ndst, N);
  // layer-1 aggregation (wave per edge)
  k_scatter1<<<CDIV(E * 32, B), B, 0, stream>>>(ei, x, nsrc, agg, E);
  // GEMM1 + bias + relu via f32 WMMA (wave per 16x16 tile)
  int Mtiles = CDIV(N, 16);
  int waves  = Mtiles * 12;
  k_gemm1<<<CDIV(waves * 32, B), B, 0, stream>>>(agg, ndst, W1, b1, score, N, Mtiles);
  // per-node scalar s1, then scalar edge aggregation
  k_s1<<<CDIV(N * 32, B), B, 0, stream>>>(score, W2, nsrc, s1, N);
  k_scatter2<<<CDIV(E, B), B, 0, stream>>>(ei, s1, sacc, E);
  // |score| -> monotone keys
  k_keys<<<CDIV(N, B), B, 0, stream>>>(sacc, ndst, b2, keys, N);
  // exact top-K threshold (2-level radix select)
  k_hist_hi<<<CDIV(N, B), B, 0, stream>>>(keys, hist1, N);
  k_sel1<<<1, 1024, 0, stream>>>(hist1, (unsigned)K, ctrl);
  k_hist_lo<<<CDIV(N, B), B, 0, stream>>>(keys, ctrl, hist2, N);
  k_sel2<<<1, 1024, 0, stream>>>(hist2, ctrl);
  // deterministic stable compaction (node-id order)
  int nb = CDIV(N, 1024);
  k_scan1<<<nb, 256, 0, stream>>>(keys, ctrl, pref, btot, N);
  k_scan2<<<1, 1, 0, stream>>>(btot, nb);
  k_emit<<<CDIV(N, B), B, 0, stream>>>(keys, ctrl, pref, btot, kept, N);
  // gather selected rows
  k_gather<<<CDIV(K * 32, B), B, 0, stream>>>(x, kept, out, K);
}